// impactDetect_74337293959193
// MI455X (gfx1250) — compile-verified
//
#include <hip/hip_runtime.h>

// Problem constants (match reference)
#define NN 50000
#define EE 800000
#define DD 64
#define TT 8000
#define CCn 8000

typedef __attribute__((ext_vector_type(2))) float v2f;
typedef __attribute__((ext_vector_type(8))) float v8f;

// ---------------- utility kernels ----------------

__global__ void zero_f32(float* __restrict__ p, int n) {
    int i = blockIdx.x * blockDim.x + threadIdx.x;
    if (i < n) p[i] = 0.0f;
}

__global__ void degree_count(const int* __restrict__ dst, float* __restrict__ cnt, int nE) {
    int e = blockIdx.x * blockDim.x + threadIdx.x;
    if (e < nE) atomicAdd(&cnt[dst[e]], 1.0f);
}

__global__ void finalize_inv(float* __restrict__ p, int n) {
    int i = blockIdx.x * blockDim.x + threadIdx.x;
    if (i < n) p[i] = 1.0f / fmaxf(p[i], 1.0f);
}

// 16 threads per edge, 4 floats each: S[dst] += X[src]
__global__ void scatter_sum(const float* __restrict__ X, const int* __restrict__ src,
                            const int* __restrict__ dst, float* __restrict__ S, int nE) {
    int t = blockIdx.x * blockDim.x + threadIdx.x;
    int e = t >> 4;
    if (e >= nE) return;
    int c = (t & 15) * 4;
    int si = src[e], di = dst[e];
    const float4 v = *(const float4*)(X + (size_t)si * DD + c);
    float* p = S + (size_t)di * DD + c;
    atomicAdd(p + 0, v.x);
    atomicAdd(p + 1, v.y);
    atomicAdd(p + 2, v.z);
    atomicAdd(p + 3, v.w);
}

// ---------------- fused SAGE layer GEMM (WMMA f32 16x16x4) ----------------
// O[n,64] = act( (S .* inv_row) @ Wl + bl + X @ Wr )
// Block = 128 threads = 4 waves; wave w owns output cols [16w,16w+16); block
// owns 16 rows. K loop of 4 per v_wmma_f32_16x16x4_f32.
//
// fp32 WMMA fragment layout (ISA 7.12.2):
//   A 16x4: lane m=lane&15 holds row m; VGPR pair holds K = {kb, kb+1},
//           kb = 2*(lane>>4).  B 4x16 mirrored (lane = N column).
//   C/D 16x16 f32: 8 VGPRs; VGPR v -> M = v + 8*(lane>>4), N = lane&15.
__global__ __launch_bounds__(128) void sage_gemm(
        const float* __restrict__ S, const float* __restrict__ inv,
        const float* __restrict__ X,
        const float* __restrict__ Wl, const float* __restrict__ bl,
        const float* __restrict__ Wr,
        float* __restrict__ O, int n, int do_relu) {
    const int lane = threadIdx.x & 31;
    const int wave = threadIdx.x >> 5;          // 0..3 -> column tile
    const int row0 = blockIdx.x * 16;
    const int col0 = wave * 16;
    const int m    = lane & 15;
    const int kb   = (lane >> 4) * 2;           // 0 or 2

    int row = row0 + m;
    int rc  = (row < n) ? row : 0;              // clamp (padded rows masked at store)
    const float rinv = inv[rc];
    const size_t rbase = (size_t)rc * DD;

    v8f acc = {};

    // C += (S .* inv) @ Wl
#pragma unroll
    for (int kk = 0; kk < DD; kk += 4) {
        v2f a, b;
        a.x = S[rbase + kk + kb]     * rinv;
        a.y = S[rbase + kk + kb + 1] * rinv;
        b.x = Wl[(kk + kb)     * DD + col0 + m];
        b.y = Wl[(kk + kb + 1) * DD + col0 + m];
        acc = __builtin_amdgcn_wmma_f32_16x16x4_f32(
                  false, a, false, b, (short)0, acc, false, false);
    }
    // C += X @ Wr
#pragma unroll
    for (int kk = 0; kk < DD; kk += 4) {
        v2f a, b;
        a.x = X[rbase + kk + kb];
        a.y = X[rbase + kk + kb + 1];
        b.x = Wr[(kk + kb)     * DD + col0 + m];
        b.y = Wr[(kk + kb + 1) * DD + col0 + m];
        acc = __builtin_amdgcn_wmma_f32_16x16x4_f32(
                  false, a, false, b, (short)0, acc, false, false);
    }

    const float bias = bl[col0 + m];
#pragma unroll
    for (int v = 0; v < 8; ++v) {
        int r = row0 + v + 8 * (lane >> 4);
        if (r < n) {
            float val = acc[v] + bias;
            if (do_relu) val = fmaxf(val, 0.0f);
            O[(size_t)r * DD + col0 + m] = val;
        }
    }
}

// ---------------- 64->1 head: out[i] = lrelu(dot(H[row],w) + b) ----------------
// row = mask ? mask[sel ? sel[i] : i] : i
__global__ void head_kernel(const float* __restrict__ H, const int* __restrict__ mask,
                            const int* __restrict__ sel, const float* __restrict__ w,
                            const float* __restrict__ bptr, float* __restrict__ out, int n) {
    int i = blockIdx.x * blockDim.x + threadIdx.x;
    if (i >= n) return;
    int r = i;
    if (sel)  r = sel[r];
    if (mask) r = mask[r];
    const float* h = H + (size_t)r * DD;
    float acc = bptr[0];
#pragma unroll
    for (int k = 0; k < DD; ++k) acc = fmaf(h[k], w[k], acc);
    out[i] = (acc > 0.0f) ? acc : 0.01f * acc;
}

// ---------------- driver ----------------

extern "C" void kernel_launch(void* const* d_in, const int* in_sizes, int n_in,
                              void* d_out, int out_size, void* d_ws, size_t ws_size,
                              hipStream_t stream) {
    const float* x       = (const float*)d_in[0];
    const float* fake_x  = (const float*)d_in[1];
    const int*   ei      = (const int*)d_in[2];   // [2,E]
    const int*   fei     = (const int*)d_in[3];   // [2,E]
    const int*   mask    = (const int*)d_in[4];
    const int*   tidx    = (const int*)d_in[5];
    const int*   cidx    = (const int*)d_in[6];
    const float* W1l = (const float*)d_in[7];
    const float* b1l = (const float*)d_in[8];
    const float* W1r = (const float*)d_in[9];
    const float* W2l = (const float*)d_in[10];
    const float* b2l = (const float*)d_in[11];
    const float* W2r = (const float*)d_in[12];
    const float* Wy1 = (const float*)d_in[13];
    const float* by1 = (const float*)d_in[14];
    const float* Wy0 = (const float*)d_in[15];
    const float* by0 = (const float*)d_in[16];
    const float* Wb  = (const float*)d_in[17];
    const float* bb  = (const float*)d_in[18];
    const float* Wp  = (const float*)d_in[19];
    const float* bp  = (const float*)d_in[20];

    const int* src_r = ei;            const int* dst_r = ei + EE;
    const int* src_f = fei;           const int* dst_f = fei + EE;

    // Workspace layout (floats): s | h1 | xZ2 | xfZ2 | invR | invF
    float* ws   = (float*)d_ws;
    float* s    = ws;                       // N*64
    float* h1   = s    + (size_t)NN * DD;   // N*64
    float* xZ2  = h1   + (size_t)NN * DD;   // N*64
    float* xfZ2 = xZ2  + (size_t)NN * DD;   // N*64
    float* invR = xfZ2 + (size_t)NN * DD;   // N
    float* invF = invR + NN;                // N  (contiguous with invR)

    float* out  = (float*)d_out;
    float* o_y1    = out;                   // T
    float* o_yc0   = out + TT;              // T
    float* o_y0    = out + 2 * TT;          // C
    float* o_yc1   = out + 2 * TT + CCn;    // C
    float* o_fp    = out + 2 * TT + 2 * CCn;            // N
    float* o_fpf   = o_fp + NN;                          // N
    float* o_tp    = o_fpf + NN;                         // N

    const int ND = NN * DD;
    dim3 blk(256);
    auto grid1 = [](int n) { return dim3((n + 255) / 256); };
    const dim3 gscat((EE * 16 + 255) / 256);
    const dim3 ggemm(NN / 16);     // 50000/16 = 3125 exactly
    const dim3 bgemm(128);

    // --- degrees (shared by both layers of each graph) ---
    zero_f32<<<grid1(2 * NN), blk, 0, stream>>>(invR, 2 * NN);
    degree_count<<<grid1(EE), blk, 0, stream>>>(dst_r, invR, EE);
    degree_count<<<grid1(EE), blk, 0, stream>>>(dst_f, invF, EE);
    finalize_inv<<<grid1(2 * NN), blk, 0, stream>>>(invR, 2 * NN);

    // --- real graph: layer 1 ---
    zero_f32<<<grid1(ND), blk, 0, stream>>>(s, ND);
    scatter_sum<<<gscat, blk, 0, stream>>>(x, src_r, dst_r, s, EE);
    sage_gemm<<<ggemm, bgemm, 0, stream>>>(s, invR, x, W1l, b1l, W1r, h1, NN, 1);
    // --- real graph: layer 2 ---
    zero_f32<<<grid1(ND), blk, 0, stream>>>(s, ND);
    scatter_sum<<<gscat, blk, 0, stream>>>(h1, src_r, dst_r, s, EE);
    sage_gemm<<<ggemm, bgemm, 0, stream>>>(s, invR, h1, W2l, b2l, W2r, xZ2, NN, 0);

    // --- fake graph: layer 1 ---
    zero_f32<<<grid1(ND), blk, 0, stream>>>(s, ND);
    scatter_sum<<<gscat, blk, 0, stream>>>(fake_x, src_f, dst_f, s, EE);
    sage_gemm<<<ggemm, bgemm, 0, stream>>>(s, invF, fake_x, W1l, b1l, W1r, h1, NN, 1);
    // --- fake graph: layer 2 ---
    zero_f32<<<grid1(ND), blk, 0, stream>>>(s, ND);
    scatter_sum<<<gscat, blk, 0, stream>>>(h1, src_f, dst_f, s, EE);
    sage_gemm<<<ggemm, bgemm, 0, stream>>>(s, invF, h1, W2l, b2l, W2r, xfZ2, NN, 0);

    // --- heads ---
    // y1  = lrelu(xZ2 [mask[treat]]  @ Wy1 + by1)
    head_kernel<<<grid1(TT),  blk, 0, stream>>>(xZ2,  mask, tidx, Wy1, by1, o_y1,  TT);
    // yc0 = lrelu(xfZ2[mask[treat]]  @ Wy1 + by1)
    head_kernel<<<grid1(TT),  blk, 0, stream>>>(xfZ2, mask, tidx, Wy1, by1, o_yc0, TT);
    // y0  = lrelu(xZ2 [mask[control]]@ Wy0 + by0)
    head_kernel<<<grid1(CCn), blk, 0, stream>>>(xZ2,  mask, cidx, Wy0, by0, o_y0,  CCn);
    // yc1 = lrelu(xfZ2[mask[control]]@ Wy0 + by0)
    head_kernel<<<grid1(CCn), blk, 0, stream>>>(xfZ2, mask, cidx, Wy0, by0, o_yc1, CCn);
    // fprob / fprob_f / treat_prob over all N rows
    head_kernel<<<grid1(NN),  blk, 0, stream>>>(xZ2,  nullptr, nullptr, Wp, bp, o_fp,  NN);
    head_kernel<<<grid1(NN),  blk, 0, stream>>>(xfZ2, nullptr, nullptr, Wp, bp, o_fpf, NN);
    head_kernel<<<grid1(NN),  blk, 0, stream>>>(xZ2,  nullptr, nullptr, Wb, bb, o_tp,  NN);
}